// Encoder_28269474742473
// MI455X (gfx1250) — compile-verified
//
#include <hip/hip_runtime.h>
#include <hip/hip_bf16.h>

#define NN 50000
#define NE 800000
#define DD 128
#define NL 5

#define ROWS 80                 // rows per block = 5 tiles of 16;  50000 = 625 * 80
#define RT   5                  // row tiles per wave
#define LSTR 132                // LDS row stride (floats): %4==0 for float4, conflict-free

static_assert(NN % ROWS == 0, "tile exact");

typedef float v2f __attribute__((ext_vector_type(2)));
typedef float v8f __attribute__((ext_vector_type(8)));

// ---------------- copy h1 = x_in (float4 vectorized) ----------------
__global__ __launch_bounds__(256) void copy4_kernel(float4* __restrict__ dst,
                                                    const float4* __restrict__ src,
                                                    long n4) {
    long i = (long)blockIdx.x * 256 + threadIdx.x;
    if (i < n4) dst[i] = src[i];
}

// ---------------- scatter: h1[dst] += x_in[src]  (2 edges / block) ----------------
__global__ __launch_bounds__(256) void scatter_kernel(float* __restrict__ h1,
                                                      const float* __restrict__ xin,
                                                      const int* __restrict__ esrc,
                                                      const int* __restrict__ edst) {
    int e = blockIdx.x * 2 + (threadIdx.x >> 7);
    int d = threadIdx.x & 127;
    long s = esrc[e];
    long t = edst[e];
    unsafeAtomicAdd(&h1[t * DD + d], xin[s * DD + d]);
}

// ---------------- fused 2-layer MLP: hpre = ReLU(ReLU(h1@W1+b1)@W2+b2) ----------------
// 1 block = 80 rows (5 row-tiles); 8 waves, wave w owns output columns [16w,16w+16).
// Inner loop: load one B fragment of W, feed 5 WMMAs (5 row tiles) -> 5x W reuse.
// Uses V_WMMA_F32_16X16X4_F32 (fp32, exact reference precision): 320 WMMA / wave.
__global__ __launch_bounds__(256) void gin_mlp_kernel(const float* __restrict__ h1,
                                                      const float* __restrict__ W1,
                                                      const float* __restrict__ b1,
                                                      const float* __restrict__ W2,
                                                      const float* __restrict__ b2,
                                                      float* __restrict__ hpre) {
    __shared__ float As[ROWS * LSTR];   // input tile  [M=80][K=128]
    __shared__ float Hs[ROWS * LSTR];   // hidden tile [M=80][K=128]

    const int tid  = threadIdx.x;
    const int wave = tid >> 5;
    const int lane = tid & 31;
    const int r    = lane & 15;
    const int half = lane >> 4;
    const int n    = (wave << 4) + r;          // output column this lane owns
    const long rowBase = (long)blockIdx.x * ROWS;

    // stage 80x128 activation tile (contiguous in global; float4, coalesced)
    {
        const float4* src4 = (const float4*)(h1 + rowBase * DD);
        for (int idx = tid; idx < ROWS * DD / 4; idx += 256) {
            int rr = idx >> 5;              // 32 float4 per row
            int cc = (idx & 31) << 2;
            *(float4*)&As[rr * LSTR + cc] = src4[idx];
        }
    }
    __syncthreads();

    // ---- GEMM1: H2 = ReLU(A @ W1 + b1) ----
    {
        v8f c[RT] = {};
        for (int k0 = 0; k0 < DD; k0 += 4) {
            int kk = k0 + 2 * half;             // K index for vgpr0 of this lane half
            v2f b;
            b.x = W1[kk * DD + n];
            b.y = W1[(kk + 1) * DD + n];
#pragma unroll
            for (int t = 0; t < RT; ++t) {
                v2f a;
                a.x = As[(t * 16 + r) * LSTR + kk];
                a.y = As[(t * 16 + r) * LSTR + kk + 1];
                c[t] = __builtin_amdgcn_wmma_f32_16x16x4_f32(false, a, false, b,
                                                             (short)0, c[t],
                                                             false, false);
            }
        }
        float bias1 = b1[n];
#pragma unroll
        for (int t = 0; t < RT; ++t)
#pragma unroll
            for (int j = 0; j < 8; ++j) {
                float v = fmaxf(c[t][j] + bias1, 0.0f);   // (M = t*16 + j + 8*half, N = n)
                Hs[(t * 16 + j + 8 * half) * LSTR + n] = v;
            }
    }
    __syncthreads();

    // ---- GEMM2: out = ReLU(H2 @ W2 + b2) ----
    {
        v8f c[RT] = {};
        for (int k0 = 0; k0 < DD; k0 += 4) {
            int kk = k0 + 2 * half;
            v2f b;
            b.x = W2[kk * DD + n];
            b.y = W2[(kk + 1) * DD + n];
#pragma unroll
            for (int t = 0; t < RT; ++t) {
                v2f a;
                a.x = Hs[(t * 16 + r) * LSTR + kk];
                a.y = Hs[(t * 16 + r) * LSTR + kk + 1];
                c[t] = __builtin_amdgcn_wmma_f32_16x16x4_f32(false, a, false, b,
                                                             (short)0, c[t],
                                                             false, false);
            }
        }
        float bias2 = b2[n];
#pragma unroll
        for (int t = 0; t < RT; ++t)
#pragma unroll
            for (int j = 0; j < 8; ++j) {
                float v = fmaxf(c[t][j] + bias2, 0.0f);
                hpre[(rowBase + t * 16 + j + 8 * half) * DD + n] = v;
            }
    }
}

// ---------------- BN: zero stats ----------------
__global__ void zero_stats_kernel(float* __restrict__ sums, float* __restrict__ sumsq) {
    sums[threadIdx.x] = 0.0f;
    sumsq[threadIdx.x] = 0.0f;
}

// ---------------- BN: per-column partial sums (coalesced rows) ----------------
__global__ __launch_bounds__(256) void bn_partial_kernel(const float* __restrict__ hpre,
                                                         float* __restrict__ sums,
                                                         float* __restrict__ sumsq) {
    __shared__ float s1[256];
    __shared__ float s2[256];
    int tid = threadIdx.x;
    int d = tid & 127;
    int ph = tid >> 7;
    float s = 0.0f, q = 0.0f;
    for (long row = (long)blockIdx.x * 2 + ph; row < NN; row += (long)gridDim.x * 2) {
        float v = hpre[row * DD + d];
        s += v;
        q += v * v;
    }
    s1[tid] = s;
    s2[tid] = q;
    __syncthreads();
    if (tid < 128) {
        unsafeAtomicAdd(&sums[d], s1[tid] + s1[tid + 128]);
        unsafeAtomicAdd(&sumsq[d], s2[tid] + s2[tid + 128]);
    }
}

// ---------------- BN: finalize -> scale/shift ----------------
__global__ void bn_finalize_kernel(const float* __restrict__ sums,
                                   const float* __restrict__ sumsq,
                                   const float* __restrict__ gamma,
                                   const float* __restrict__ beta,
                                   float* __restrict__ scale,
                                   float* __restrict__ shift) {
    int d = threadIdx.x;
    const float invN = 1.0f / (float)NN;
    float mu = sums[d] * invN;
    float var = sumsq[d] * invN - mu * mu;
    float rstd = rsqrtf(var + 1e-5f);
    float sc = rstd * gamma[d];
    scale[d] = sc;
    shift[d] = beta[d] - mu * sc;
}

// ---------------- BN: apply (writes xs[i], optionally final h) ----------------
__global__ __launch_bounds__(256) void bn_apply_kernel(const float* __restrict__ hpre,
                                                       const float* __restrict__ scale,
                                                       const float* __restrict__ shift,
                                                       float* __restrict__ out1,
                                                       float* __restrict__ out2) {
    long i = (long)blockIdx.x * 256 + threadIdx.x;
    if (i >= (long)NN * DD) return;
    int d = (int)(i & 127);
    float v = hpre[i] * scale[d] + shift[d];
    out1[i] = v;
    if (out2) out2[i] = v;
}

extern "C" void kernel_launch(void* const* d_in, const int* in_sizes, int n_in,
                              void* d_out, int out_size, void* d_ws, size_t ws_size,
                              hipStream_t stream) {
    const float* x     = (const float*)d_in[0];
    const int*   eidx  = (const int*)d_in[1];   // [2][NE]; batch (d_in[2]) unused
    const float* W1    = (const float*)d_in[3];
    const float* b1    = (const float*)d_in[4];
    const float* W2    = (const float*)d_in[5];
    const float* b2    = (const float*)d_in[6];
    const float* gamma = (const float*)d_in[7];
    const float* beta  = (const float*)d_in[8];

    float* out = (float*)d_out;
    float* xs  = out + (long)NN * DD;           // stacked per-layer outputs

    float* ws    = (float*)d_ws;
    float* h1    = ws;                          // [NN*DD]
    float* hpre  = h1 + (long)NN * DD;          // [NN*DD]
    float* sums  = hpre + (long)NN * DD;        // [128]
    float* sumsq = sums + 128;                  // [128]
    float* scale = sumsq + 128;                 // [128]
    float* shift = scale + 128;                 // [128]

    const int* esrc = eidx;
    const int* edst = eidx + NE;

    const long ND = (long)NN * DD;

    for (int i = 0; i < NL; ++i) {
        const float* in = (i == 0) ? x : (xs + (long)(i - 1) * ND);

        copy4_kernel<<<(int)(ND / 4 / 256), 256, 0, stream>>>(
            (float4*)h1, (const float4*)in, ND / 4);

        scatter_kernel<<<NE / 2, 256, 0, stream>>>(h1, in, esrc, edst);

        gin_mlp_kernel<<<NN / ROWS, 256, 0, stream>>>(
            h1, W1 + (long)i * DD * DD, b1 + i * DD,
            W2 + (long)i * DD * DD, b2 + i * DD, hpre);

        zero_stats_kernel<<<1, 128, 0, stream>>>(sums, sumsq);
        bn_partial_kernel<<<128, 256, 0, stream>>>(hpre, sums, sumsq);
        bn_finalize_kernel<<<1, 128, 0, stream>>>(sums, sumsq,
                                                  gamma + i * DD, beta + i * DD,
                                                  scale, shift);
        bn_apply_kernel<<<(int)(ND / 256), 256, 0, stream>>>(
            hpre, scale, shift, xs + (long)i * ND,
            (i == NL - 1) ? out : nullptr);
    }
}